// AttentiveTransformer_16501264351727
// MI455X (gfx1250) — compile-verified
//
#include <hip/hip_runtime.h>
#include <math.h>

typedef float v2f __attribute__((ext_vector_type(2)));
typedef float v8f __attribute__((ext_vector_type(8)));

#define B_ROWS 131072
#define D_IN   128
#define UNITS  512

// ---- GEMM phase geometry ----
#define NWG_A        1024
#define TILES_PER_WG ((B_ROWS / 16) / NWG_A)   // 8 sixteen-row tiles per WG
// LDS layout: W stored as interleaved (even-K, odd-K) pairs, one "row" per K-pair.
//   Wl[kp*WP + 2*n + s] = W[2*kp + s][n]
// WP = 2*512 + 32 pad  -> lanes 0-15 and 16-31 of a b64 fragment read hit all 64 banks.
#define WP 1056
// A tile 16x128 with pitch 132 (528B: 16B aligned, lane bank stride 4 -> conflict free)
#define AP 132
#define LDS_A_OFF (64 * WP)
#define LDS_BYTES ((64 * WP + 16 * AP) * 4)    // 278784 B < 320 KB WGP LDS

// ---- sparsemax phase geometry ----
#define NWG_C         1024
#define ROWS_PER_WAVE 16                        // 1024 WG * 8 waves * 16 rows = 131072

// =====================================================================
// Phase A: x = inputs @ W  (fp32 WMMA 16x16x4), plus per-WG column
// sum / sum-of-squares partials for the batch-norm statistics.
// =====================================================================
__global__ void __launch_bounds__(256, 1)
tabnet_gemm_stats(const float* __restrict__ inp, const float* __restrict__ W,
                  float* __restrict__ x, float* __restrict__ psum,
                  float* __restrict__ psumsq)
{
    extern __shared__ float smem[];
    float* Wl = smem;
    float* Al = smem + LDS_A_OFF;

    const int tid  = threadIdx.x;
    const int wave = tid >> 5;
    const int lane = tid & 31;

    // Stage W (128x512 f32 = 256KB, L2-resident) into LDS, interleaving K pairs
    // so each lane's WMMA B-fragment is a single ds_load_b64.
    for (int p = tid; p < 64 * UNITS; p += 256) {
        const int kp = p >> 9;             // K/2
        const int n  = p & (UNITS - 1);
        Wl[kp * WP + 2 * n + 0] = W[(2 * kp + 0) * UNITS + n];
        Wl[kp * WP + 2 * n + 1] = W[(2 * kp + 1) * UNITS + n];
    }

    const int nbase = wave * 64;           // 64 output columns per wave
    const int rowA  = lane & 15;
    const int khalf = lane >> 4;           // 0: K+0/K+1, 1: K+2/K+3
    const int koffA = khalf * 2;
    const int ncol  = nbase + (lane & 15);

    float colsum[4] = {0.f, 0.f, 0.f, 0.f};
    float colsq[4]  = {0.f, 0.f, 0.f, 0.f};

    for (int t = 0; t < TILES_PER_WG; ++t) {
        const int row0 = (blockIdx.x * TILES_PER_WG + t) * 16;
        __syncthreads();  // Al reuse fence (covers W staging on t==0)
        {   // cooperative, coalesced 16x128 A-tile load (8 KB)
            const float4* src = (const float4*)(inp + (size_t)row0 * D_IN);
            for (int i = tid; i < (16 * D_IN) / 4; i += 256) {
                const int r = i >> 5, c4 = i & 31;
                *(float4*)(Al + r * AP + c4 * 4) = src[i];
            }
        }
        if (t + 1 < TILES_PER_WG)  // global_prefetch_b8 next tile
            __builtin_prefetch(inp + ((size_t)row0 + 16) * D_IN + tid * 8, 0, 1);
        __syncthreads();

        v8f acc[4] = {};
        #pragma unroll 4
        for (int kk = 0; kk < D_IN; kk += 4) {
            // A fragment (16x4 f32): one ds_load_b64 per lane
            const v2f a = *(const v2f*)(Al + rowA * AP + kk + koffA);
            const int kp = (kk >> 1) + khalf;
            const float* wb = Wl + kp * WP + 2 * ncol;
            const v2f b0 = *(const v2f*)(wb + 0);    // cols nbase+ 0..15
            const v2f b1 = *(const v2f*)(wb + 32);   // cols nbase+16..31
            const v2f b2 = *(const v2f*)(wb + 64);   // cols nbase+32..47
            const v2f b3 = *(const v2f*)(wb + 96);   // cols nbase+48..63
            acc[0] = __builtin_amdgcn_wmma_f32_16x16x4_f32(false, a, false, b0,
                                                           (short)0, acc[0], false, false);
            acc[1] = __builtin_amdgcn_wmma_f32_16x16x4_f32(false, a, false, b1,
                                                           (short)0, acc[1], false, false);
            acc[2] = __builtin_amdgcn_wmma_f32_16x16x4_f32(false, a, false, b2,
                                                           (short)0, acc[2], false, false);
            acc[3] = __builtin_amdgcn_wmma_f32_16x16x4_f32(false, a, false, b3,
                                                           (short)0, acc[3], false, false);
        }

        // C layout: VGPR j, lanes 0-15 -> row j, lanes 16-31 -> row j+8, col = lane%16
        #pragma unroll
        for (int c = 0; c < 4; ++c) {
            const int col = nbase + c * 16 + (lane & 15);
            float* xout = x + (size_t)(row0 + khalf * 8) * UNITS + col;
            #pragma unroll
            for (int j = 0; j < 8; ++j) {
                const float v = acc[c][j];
                xout[(size_t)j * UNITS] = v;
                colsum[c] += v;
                colsq[c]   = fmaf(v, v, colsq[c]);
            }
        }
    }

    // deterministic partials: 2 entries (row-halves) per column per WG
    #pragma unroll
    for (int c = 0; c < 4; ++c) {
        const int idx = (nbase + c * 16 + (lane & 15)) * 2 + khalf;
        psum  [(size_t)blockIdx.x * (2 * UNITS) + idx] = colsum[c];
        psumsq[(size_t)blockIdx.x * (2 * UNITS) + idx] = colsq[c];
    }
}

// =====================================================================
// Phase B: reduce partials -> fused BN scale/shift per column.
// =====================================================================
__global__ void __launch_bounds__(512)
tabnet_bn_reduce(const float* __restrict__ psum, const float* __restrict__ psumsq,
                 const float* __restrict__ gamma, const float* __restrict__ beta,
                 float* __restrict__ scsh)
{
    const int c = threadIdx.x;  // 0..511
    float s = 0.f, ss = 0.f;
    for (int w = 0; w < NWG_A; ++w) {
        const float* ps = psum   + (size_t)w * (2 * UNITS);
        const float* pq = psumsq + (size_t)w * (2 * UNITS);
        s  += ps[2 * c] + ps[2 * c + 1];
        ss += pq[2 * c] + pq[2 * c + 1];
    }
    const float invB  = 1.0f / (float)B_ROWS;
    const float mean  = s * invB;
    const float var   = ss * invB - mean * mean;
    const float scale = gamma[c] * rsqrtf(var + 1e-3f);
    scsh[c]         = scale;
    scsh[c + UNITS] = beta[c] - mean * scale;
}

// =====================================================================
// Phase C: z = prior * BN(x); mask = sparsemax(z) (sort-free Newton on the
// simplex-projection threshold); entropy partial per WG. One wave per row.
// =====================================================================
__global__ void __launch_bounds__(256)
tabnet_sparsemax(const float* __restrict__ prior, const float* __restrict__ scsh,
                 float* __restrict__ xio, float* __restrict__ entp)
{
    __shared__ float sent[8];
    const int tid = threadIdx.x, wave = tid >> 5, lane = tid & 31;

    float scv[16], shv[16];
    #pragma unroll
    for (int i = 0; i < 16; i += 4) {
        const float4 a = *(const float4*)(scsh + lane * 16 + i);
        const float4 b = *(const float4*)(scsh + UNITS + lane * 16 + i);
        scv[i] = a.x; scv[i + 1] = a.y; scv[i + 2] = a.z; scv[i + 3] = a.w;
        shv[i] = b.x; shv[i + 1] = b.y; shv[i + 2] = b.z; shv[i + 3] = b.w;
    }

    float eacc = 0.f;
    for (int r = 0; r < ROWS_PER_WAVE; ++r) {
        const size_t row = ((size_t)blockIdx.x * 8 + wave) * ROWS_PER_WAVE + r;
        const float* xr = xio   + row * UNITS + lane * 16;
        const float* pr = prior + row * UNITS + lane * 16;

        float z[16];
        #pragma unroll
        for (int i = 0; i < 16; i += 4) {
            const float4 xv = *(const float4*)(xr + i);
            const float4 pv = *(const float4*)(pr + i);
            z[i + 0] = pv.x * fmaf(xv.x, scv[i + 0], shv[i + 0]);
            z[i + 1] = pv.y * fmaf(xv.y, scv[i + 1], shv[i + 1]);
            z[i + 2] = pv.z * fmaf(xv.z, scv[i + 2], shv[i + 2]);
            z[i + 3] = pv.w * fmaf(xv.w, scv[i + 3], shv[i + 3]);
        }

        float mx = z[0];
        #pragma unroll
        for (int i = 1; i < 16; ++i) mx = fmaxf(mx, z[i]);
        #pragma unroll
        for (int o = 16; o >= 1; o >>= 1) mx = fmaxf(mx, __shfl_xor(mx, o, 32));

        // Newton on f(tau)=sum(max(z-tau,0))-1 : convex piecewise-linear,
        // monotone convergence from tau0 = max-1 (k >= 1 always).
        float tau = mx - 1.0f;
        for (int it = 0; it < 64; ++it) {
            float s = 0.f, kf = 0.f;
            #pragma unroll
            for (int i = 0; i < 16; ++i) {
                const float d = z[i] - tau;
                if (d > 0.f) { s += d; kf += 1.f; }
            }
            #pragma unroll
            for (int o = 16; o >= 1; o >>= 1) {
                s  += __shfl_xor(s,  o, 32);
                kf += __shfl_xor(kf, o, 32);
            }
            const float step = (s - 1.0f) / kf;
            tau += step;
            if (!(step > 1e-7f)) break;   // wave-uniform condition
        }

        float e = 0.f;
        float* outr = xio + row * UNITS + lane * 16;
        #pragma unroll
        for (int i = 0; i < 16; i += 4) {
            float4 m;
            m.x = fmaxf(z[i + 0] - tau, 0.f);
            m.y = fmaxf(z[i + 1] - tau, 0.f);
            m.z = fmaxf(z[i + 2] - tau, 0.f);
            m.w = fmaxf(z[i + 3] - tau, 0.f);
            e -= m.x * __logf(m.x + 1e-15f);   // m==0 contributes exactly 0
            e -= m.y * __logf(m.y + 1e-15f);
            e -= m.z * __logf(m.z + 1e-15f);
            e -= m.w * __logf(m.w + 1e-15f);
            *(float4*)(outr + i) = m;          // in-place mask write
        }
        #pragma unroll
        for (int o = 16; o >= 1; o >>= 1) e += __shfl_xor(e, o, 32);
        if (lane == 0) eacc += e;
    }

    if (lane == 0) sent[wave] = eacc;
    __syncthreads();
    if (tid == 0) {
        float t = 0.f;
        #pragma unroll
        for (int w = 0; w < 8; ++w) t += sent[w];
        entp[blockIdx.x] = t;
    }
}

// =====================================================================
// Phase D: deterministic tree reduction of entropy partials -> loss.
// =====================================================================
__global__ void __launch_bounds__(256)
tabnet_loss(const float* __restrict__ entp, float* __restrict__ lossp)
{
    __shared__ float sb[256];
    const int t = threadIdx.x;
    float v = 0.f;
    for (int i = t; i < NWG_C; i += 256) v += entp[i];
    sb[t] = v;
    __syncthreads();
    for (int o = 128; o >= 1; o >>= 1) {
        if (t < o) sb[t] += sb[t + o];
        __syncthreads();
    }
    if (t == 0)
        lossp[0] = sb[0] * (1.0f / (float)B_ROWS) * (1.0e-3f / 3.0f);
}

extern "C" void kernel_launch(void* const* d_in, const int* in_sizes, int n_in,
                              void* d_out, int out_size, void* d_ws, size_t ws_size,
                              hipStream_t stream)
{
    (void)in_sizes; (void)n_in; (void)out_size; (void)ws_size;
    const float* inp   = (const float*)d_in[0];
    const float* prior = (const float*)d_in[1];
    const float* W     = (const float*)d_in[2];
    const float* gamma = (const float*)d_in[3];
    const float* beta  = (const float*)d_in[4];

    float* out  = (float*)d_out;
    float* x    = out;                               // mask region doubles as x scratch
    float* loss = out + (size_t)B_ROWS * UNITS;

    float* ws     = (float*)d_ws;
    float* psum   = ws;                              // NWG_A * 1024 floats
    float* psumsq = psum   + (size_t)NWG_A * (2 * UNITS);
    float* scsh   = psumsq + (size_t)NWG_A * (2 * UNITS);  // 1024 floats
    float* entp   = scsh + 2 * UNITS;                // NWG_C floats

    tabnet_gemm_stats<<<NWG_A, 256, LDS_BYTES, stream>>>(inp, W, x, psum, psumsq);
    tabnet_bn_reduce<<<1, 512, 0, stream>>>(psum, psumsq, gamma, beta, scsh);
    tabnet_sparsemax<<<NWG_C, 256, 0, stream>>>(prior, scsh, x, entp);
    tabnet_loss<<<1, 256, 0, stream>>>(entp, loss);
}